// FADE_21449066676348
// MI455X (gfx1250) — compile-verified
//
#include <hip/hip_runtime.h>
#include <hip/hip_bf16.h>

typedef __attribute__((ext_vector_type(2))) float v2f;
typedef __attribute__((ext_vector_type(8))) float v8f;

#define B_    4
#define CDE_  256
#define CEN_  256
#define H_    64
#define W_    64
#define HO_   128
#define WO_   128
#define EMB_  64
#define KK2_  25     // 5x5 kernel taps
#define K3_   576    // 64 ch * 9 taps for the 3x3 conv GEMM

// ---------------------------------------------------------------------------
// 1x1 conv as GEMM via V_WMMA_F32_16X16X4_F32, HW compile-time so all B-operand
// loads are base+immediate with a single pointer bump per K-step.
// in:(B,256,HW) w:(64,256) out:(B,64,HW). blockDim=(32,4): ty = oc-tile.
// A lane layout (16x4 f32): lanes0-15 hold {K0,K1}, lanes16-31 hold {K2,K3}, M=lane&15.
// B lane layout (4x16 f32): VGPR0 = rows K0/K2 (per half), VGPR1 = rows K1/K3, N=lane&15.
// D lane layout: VGPR r = row (r + 8*half), col = lane&15.
// ---------------------------------------------------------------------------
template <int HW>
__global__ void conv1x1_wmma(const float* __restrict__ in, const float* __restrict__ w,
                             const float* __restrict__ bias, float* __restrict__ out) {
    const int lane = threadIdx.x;
    const int half = lane >> 4;
    const int m    = lane & 15;
    const int oc0  = threadIdx.y * 16;
    const int px   = blockIdx.x * 16 + m;
    const int b    = blockIdx.z;

    const float* p = in + (size_t)b * CDE_ * HW + (size_t)(2 * half) * HW + px;
    const float* q = w + (size_t)(oc0 + m) * CDE_ + 2 * half;

    v8f acc = {};
    #pragma unroll 8
    for (int k0 = 0; k0 < CDE_; k0 += 4) {
        v2f a = *(const v2f*)q;      // A: w[oc0+m][kk], w[oc0+m][kk+1]
        v2f bv;
        bv.x = p[0];                 // B: in[kk][px]     (immediate offset 0)
        bv.y = p[HW];                //    in[kk+1][px]   (immediate offset HW*4)
        acc = __builtin_amdgcn_wmma_f32_16x16x4_f32(false, a, false, bv,
                                                    (short)0, acc, false, false);
        p += 4 * HW;
        q += 4;
    }

    float* ob = out + ((size_t)b * EMB_ + oc0 + 8 * half) * HW + px;
    #pragma unroll
    for (int r = 0; r < 8; ++r) {
        float v = acc[r];
        if (bias) v += bias[oc0 + 8 * half + r];
        ob[(size_t)r * HW] = v;
    }
}

// ---------------------------------------------------------------------------
// Repack 3x3 content-encoder weights (25,64,3,3) -> wr[32][t*64+c], zero-padded
// to 32 output rows so the WMMA A-operand never reads out of bounds.
// ---------------------------------------------------------------------------
__global__ void repack_wce(const float* __restrict__ wce, float* __restrict__ wr) {
    for (int i = blockIdx.x * blockDim.x + threadIdx.x; i < 32 * K3_;
         i += gridDim.x * blockDim.x) {
        const int oc = i / K3_;
        const int k  = i - oc * K3_;
        const int t  = k >> 6;       // tap 0..8
        const int c  = k & 63;       // channel 0..63
        wr[i] = (oc < KK2_) ? wce[(size_t)oc * K3_ + c * 9 + t] : 0.0f;
    }
}

// ---------------------------------------------------------------------------
// 3x3 conv (64ch -> 25ch, pad=1) as tap-major GEMM (K = 9*64) via f32 WMMA.
// Outer tap loop fully unrolled (constants fold); inner c-loop is tight:
// one v2f A load + two immediate-offset B loads + cndmask border select + WMMA.
// Border handling branchless so EXEC stays all-ones across every WMMA.
// If lo != nullptr, adds nearest-upsampled low-res logits (fuses up2(+)).
// ---------------------------------------------------------------------------
template <int Hin, int Win>
__global__ void conv3x3_wmma(const float* __restrict__ in, const float* __restrict__ wr,
                             const float* __restrict__ bce, const float* __restrict__ lo,
                             float* __restrict__ out) {
    const int lane = threadIdx.x;
    const int half = lane >> 4;
    const int m    = lane & 15;
    const int oc0  = threadIdx.y * 16;     // 0 or 16
    const int Y    = blockIdx.y;
    const int X    = blockIdx.x * 16 + m;
    const int b    = blockIdx.z;

    constexpr int HWin = Hin * Win;
    const float* inb  = in + (size_t)b * EMB_ * HWin + (size_t)(2 * half) * HWin;
    const float* wrow = wr + (size_t)(oc0 + m) * K3_ + 2 * half;

    v8f acc = {};
    #pragma unroll
    for (int t = 0; t < 9; ++t) {
        const int dy = t / 3 - 1, dx = t % 3 - 1;     // compile-time constants
        const int yy = Y + dy;
        const int xx = X + dx;
        const int yyc = min(max(yy, 0), Hin - 1);
        const int xxc = min(max(xx, 0), Win - 1);
        const bool ok = ((unsigned)yy < (unsigned)Hin) & ((unsigned)xx < (unsigned)Win);
        const float* p = inb + (size_t)yyc * Win + xxc;
        const float* q = wrow + t * 64;
        #pragma unroll 4
        for (int c0 = 0; c0 < 64; c0 += 4) {
            v2f a = *(const v2f*)q;
            const float f0 = p[0];
            const float f1 = p[HWin];
            v2f bv;
            bv.x = ok ? f0 : 0.0f;
            bv.y = ok ? f1 : 0.0f;
            acc = __builtin_amdgcn_wmma_f32_16x16x4_f32(false, a, false, bv,
                                                        (short)0, acc, false, false);
            p += 4 * HWin;
            q += 4;
        }
    }

    #pragma unroll
    for (int r = 0; r < 8; ++r) {
        const int oc = oc0 + 8 * half + r;
        if (oc < KK2_) {
            float v = acc[r] + bce[oc];
            if (lo)  // nearest up2 of the low-res (de-side) logits, bias already inside
                v += lo[((size_t)(b * KK2_ + oc) * (Hin / 2) + (Y >> 1)) * (Win / 2) + (X >> 1)];
            out[((size_t)(b * KK2_ + oc) * Hin + Y) * Win + X] = v;
        }
    }
}

// ---------------------------------------------------------------------------
// Gate at low res: sigmoid(conv1x1(de, w_gate)+b_gate), sampled by >>1 later.
// ---------------------------------------------------------------------------
__global__ void gate_kernel(const float* __restrict__ de, const float* __restrict__ wg,
                            const float* __restrict__ bg, float* __restrict__ gate_lo) {
    const int idx = blockIdx.x * blockDim.x + threadIdx.x;  // b*4096 + p
    const int b = idx >> 12;
    const int p = idx & 4095;
    const float* dp = de + (size_t)b * CDE_ * (H_ * W_) + p;
    float s = bg[0];
    #pragma unroll 8
    for (int c = 0; c < CDE_; ++c) s += dp[(size_t)c * (H_ * W_)] * wg[c];
    gate_lo[idx] = 1.0f / (1.0f + __expf(-s));
}

// ---------------------------------------------------------------------------
// Fused softmax(25) + CARAFE(5x5, scale 2, 256ch) + gate blend.
// Block = (32,8): 32 output pixels of one row, softmax kernels staged in LDS,
// 8 thread-rows sweep the 256 channels (coalesced en/out access along X).
// ---------------------------------------------------------------------------
__global__ void carafe_blend(const float* __restrict__ de, const float* __restrict__ en,
                             const float* __restrict__ logits,
                             const float* __restrict__ gate_lo, float* __restrict__ out) {
    __shared__ float kw[KK2_][32];
    const int tx = threadIdx.x;            // 0..31 -> X
    const int ty = threadIdx.y;            // 0..7  -> channel phase
    const int X  = blockIdx.x * 32 + tx;
    const int Y  = blockIdx.y;
    const int b  = blockIdx.z;

    if (ty == 0) {                          // per-pixel softmax over 25 taps
        const float* lp = logits + ((size_t)b * KK2_ * HO_ + Y) * WO_ + X;
        float v[KK2_];
        float mx = -1e30f;
        #pragma unroll
        for (int q = 0; q < KK2_; ++q) { v[q] = lp[(size_t)q * HO_ * WO_]; mx = fmaxf(mx, v[q]); }
        float s = 0.0f;
        #pragma unroll
        for (int q = 0; q < KK2_; ++q) { v[q] = __expf(v[q] - mx); s += v[q]; }
        const float inv = 1.0f / s;
        #pragma unroll
        for (int q = 0; q < KK2_; ++q) kw[q][tx] = v[q] * inv;
    }
    __syncthreads();

    const int y = Y >> 1, x = X >> 1;
    const float g = gate_lo[((size_t)b * H_ + y) * W_ + x];
    const float* deb = de + (size_t)b * CDE_ * (H_ * W_);
    const float* enb = en + (size_t)b * CEN_ * (HO_ * WO_);
    float* ob = out + (size_t)b * CEN_ * (HO_ * WO_);

    for (int c = ty; c < CDE_; c += 8) {
        const float* dc = deb + (size_t)c * (H_ * W_);
        float acc = 0.0f;
        #pragma unroll
        for (int dy = 0; dy < 5; ++dy) {
            const int yy = y + dy - 2;
            const bool yok = (unsigned)yy < (unsigned)H_;
            #pragma unroll
            for (int dx = 0; dx < 5; ++dx) {
                const int xx = x + dx - 2;
                const float dv = (yok && (unsigned)xx < (unsigned)W_)
                                     ? dc[yy * W_ + xx] : 0.0f;
                acc += kw[dy * 5 + dx][tx] * dv;
            }
        }
        const size_t oi = ((size_t)c * HO_ + Y) * WO_ + X;
        ob[oi] = g * enb[oi] + (1.0f - g) * acc;
    }
}

// ---------------------------------------------------------------------------
extern "C" void kernel_launch(void* const* d_in, const int* in_sizes, int n_in,
                              void* d_out, int out_size, void* d_ws, size_t ws_size,
                              hipStream_t stream) {
    (void)in_sizes; (void)n_in; (void)out_size; (void)ws_size;
    const float* en     = (const float*)d_in[0];   // (4,256,128,128)
    const float* de     = (const float*)d_in[1];   // (4,256,64,64)
    const float* w_gate = (const float*)d_in[2];   // (1,256,1,1)
    const float* b_gate = (const float*)d_in[3];   // (1,)
    const float* w_cen  = (const float*)d_in[4];   // (64,256,1,1)
    const float* b_cen  = (const float*)d_in[5];   // (64,)
    const float* w_cde  = (const float*)d_in[6];   // (64,256,1,1)
    const float* w_ce   = (const float*)d_in[7];   // (25,64,3,3)
    const float* b_ce   = (const float*)d_in[8];   // (25,)
    float* out = (float*)d_out;

    // workspace layout (floats), total ~7.33M floats (~29.3 MB)
    float* wsf = (float*)d_ws;
    size_t o = 0;
    float* enc       = wsf + o; o += (size_t)B_ * EMB_ * HO_ * WO_;  // 4,194,304
    float* dec       = wsf + o; o += (size_t)B_ * EMB_ * H_  * W_;   // 1,048,576
    float* gate_lo   = wsf + o; o += (size_t)B_ * H_ * W_;           //    16,384
    float* lo_logits = wsf + o; o += (size_t)B_ * KK2_ * H_ * W_;    //   409,600
    float* logits    = wsf + o; o += (size_t)B_ * KK2_ * HO_ * WO_;  // 1,638,400
    float* wr        = wsf + o; o += 32 * K3_;                       //    18,432

    // weight repack for 3x3 GEMM
    repack_wce<<<dim3(72), dim3(256), 0, stream>>>(w_ce, wr);
    // 1x1 convs (WMMA f32 GEMMs)
    conv1x1_wmma<HO_ * WO_><<<dim3((HO_ * WO_) / 16, 1, B_), dim3(32, 4), 0, stream>>>(
        en, w_cen, b_cen, enc);
    conv1x1_wmma<H_ * W_><<<dim3((H_ * W_) / 16, 1, B_), dim3(32, 4), 0, stream>>>(
        de, w_cde, nullptr, dec);
    // gate (sigmoid of 1-channel 1x1 conv, low res)
    gate_kernel<<<dim3((B_ * H_ * W_) / 256), dim3(256), 0, stream>>>(
        de, w_gate, b_gate, gate_lo);
    // 3x3 convs (WMMA f32 GEMMs, K = 576); second fuses the up2(+) add
    conv3x3_wmma<H_, W_><<<dim3(W_ / 16, H_, B_), dim3(32, 2), 0, stream>>>(
        dec, wr, b_ce, nullptr, lo_logits);
    conv3x3_wmma<HO_, WO_><<<dim3(WO_ / 16, HO_, B_), dim3(32, 2), 0, stream>>>(
        enc, wr, b_ce, lo_logits, logits);
    // softmax + CARAFE + gate blend
    carafe_blend<<<dim3(WO_ / 32, HO_, B_), dim3(32, 8), 0, stream>>>(
        de, en, logits, gate_lo, out);
}